// LearnableMaskGenerator_25383256719878
// MI455X (gfx1250) — compile-verified
//
#include <hip/hip_runtime.h>
#include <math.h>

typedef __attribute__((ext_vector_type(2))) float v2f;
typedef __attribute__((ext_vector_type(8))) float v8f;

#define B_ 2
#define L_ 384
#define S_ 384
#define D_ 512
#define H_ 256

// ---------------------------------------------------------------------------
// Tiny kernel: Gq[b,h] = q_global[b]·Wqg[:,h], Gk[b,h] = k_global[b]·Wkg[:,h]
// (rank-1 global terms; 1024 dot-products of length 512 — negligible)
// ---------------------------------------------------------------------------
__global__ void proj_globals_kernel(const float* __restrict__ qg,
                                    const float* __restrict__ kg,
                                    const float* __restrict__ W1,
                                    float* __restrict__ Gq,
                                    float* __restrict__ Gk) {
    int t = blockIdx.x * blockDim.x + threadIdx.x;
    if (t >= 2 * B_ * H_) return;
    int which = t / (B_ * H_);          // 0 -> Gq, 1 -> Gk
    int idx   = t % (B_ * H_);
    int b = idx / H_, h = idx % H_;
    const float* vec = (which ? kg : qg) + b * D_;
    const float* W   = W1 + (which ? 3 : 2) * D_ * H_;  // Wqg rows 1024.., Wkg rows 1536..
    float acc = 0.f;
#pragma unroll 8
    for (int d = 0; d < D_; ++d) acc = fmaf(vec[d], W[d * H_ + h], acc);
    (which ? Gk : Gq)[idx] = acc;
}

// ---------------------------------------------------------------------------
// Tiny kernel: gw[h] = gamma[h]*W2[h];  consts[0]=Σ gamma*W2,
// consts[1]=Σ beta*W2 + b2   (LayerNorm+matvec fusion constants)
// ---------------------------------------------------------------------------
__global__ void prep_kernel(const float* __restrict__ gamma,
                            const float* __restrict__ beta,
                            const float* __restrict__ W2,
                            const float* __restrict__ b2,
                            float* __restrict__ gw,
                            float* __restrict__ consts) {
    __shared__ float s0[H_], s1[H_];
    int t = threadIdx.x;
    float w   = W2[t];
    float gwv = gamma[t] * w;
    gw[t] = gwv;
    s0[t] = gwv;
    s1[t] = beta[t] * w;
    __syncthreads();
    for (int off = H_ / 2; off > 0; off >>= 1) {
        if (t < off) { s0[t] += s0[t + off]; s1[t] += s1[t + off]; }
        __syncthreads();
    }
    if (t == 0) { consts[0] = s0[0]; consts[1] = s1[0] + b2[0]; }
}

// ---------------------------------------------------------------------------
// fp32 WMMA GEMM: Out[m,n] = Σ_k A[m,k]·Bw[k,n] + Gvec[batch(m),n] + b1scale*b1[n]
// One wave computes one 16x16 tile via 128 x V_WMMA_F32_16X16X4_F32.
// Per-lane layout (ISA 7.12.2, 32-bit operands):
//   A 16x4 : lane holds {A[m][kb], A[m][kb+1]}, m=lane&15, kb=2*(lane>>4)
//   B 4x16 : lane holds {B[kb][n], B[kb+1][n]}, n=lane&15
//   C/D    : VGPR r -> row r + 8*(lane>>4), col lane&15
// Batch index is uniform per tile (L_ % 16 == 0), so bb = m0/L_ is scalar —
// avoids per-lane magic-number division in the epilogue.
// ---------------------------------------------------------------------------
__global__ void gemm_proj_kernel(const float* __restrict__ A,    // [M, 512]
                                 const float* __restrict__ Bw,   // [512, 256] row-major
                                 const float* __restrict__ Gvec, // [B_, 256]
                                 const float* __restrict__ b1,   // [256]
                                 float b1scale,
                                 float* __restrict__ Out) {      // [M, 256]
    const int m0   = blockIdx.x * 16;
    const int n0   = blockIdx.y * 16;
    const int bb   = m0 / L_;           // uniform: tile never straddles a batch
    const int lane = threadIdx.x;
    const int r16  = lane & 15;
    const int hi   = lane >> 4;
    const float* arow = A + (m0 + r16) * D_;
    const int n = n0 + r16;
    v8f c = {};
#pragma unroll 4
    for (int kk = 0; kk < D_; kk += 4) {
        const int kb = kk + 2 * hi;
        v2f a, b;
        a.x = arow[kb];
        a.y = arow[kb + 1];
        b.x = Bw[kb * H_ + n];
        b.y = Bw[(kb + 1) * H_ + n];
        c = __builtin_amdgcn_wmma_f32_16x16x4_f32(false, a, false, b,
                                                  (short)0, c, false, false);
    }
    const float addend = Gvec[bb * H_ + n] + b1scale * b1[n];
#pragma unroll
    for (int r = 0; r < 8; ++r) {
        const int row = m0 + r + 8 * hi;
        Out[row * H_ + n] = c[r] + addend;
    }
}

// ---------------------------------------------------------------------------
// fp32 WMMA GEMM: sim[b,l,s] = (q[b,l,:]·k[b,s,:]) / sqrt(512)
// B operand is kᵀ: B[kd][n] = k[b, n0+n, kd]  (per-lane b64 load along D)
// ---------------------------------------------------------------------------
__global__ void gemm_sim_kernel(const float* __restrict__ q,
                                const float* __restrict__ k,
                                float* __restrict__ sim) {
    const int bb   = blockIdx.z;
    const int m0   = blockIdx.x * 16;
    const int n0   = blockIdx.y * 16;
    const int lane = threadIdx.x;
    const int r16  = lane & 15;
    const int hi   = lane >> 4;
    const float* A  = q + ((size_t)bb * L_ + m0 + r16) * D_;
    const float* Kr = k + ((size_t)bb * S_ + n0 + r16) * D_;
    v8f c = {};
#pragma unroll 4
    for (int kk = 0; kk < D_; kk += 4) {
        const int kb = kk + 2 * hi;
        v2f a, b;
        a.x = A[kb];  a.y = A[kb + 1];
        b.x = Kr[kb]; b.y = Kr[kb + 1];
        c = __builtin_amdgcn_wmma_f32_16x16x4_f32(false, a, false, b,
                                                  (short)0, c, false, false);
    }
    const float scale = 0.044194173824159216f; // 1/sqrt(512)
    float* srow = sim + (((size_t)bb * L_ + m0 + 8 * hi) * S_ + n0 + r16);
#pragma unroll
    for (int r = 0; r < 8; ++r) {
        srow[(size_t)r * S_] = c[r] * scale;
    }
}

// ---------------------------------------------------------------------------
// Heavy fused kernel: per (l,s) pair, H=256 GELU evals + 3 reductions,
// fused LN threshold, sim, Gumbel straight-through hard decision.
// 16x16 pair tile per 256-thread block; Hq/Hk tiles in LDS, stride 257
// (256 % 64banks == 0 would be 16-way conflicted; 257 -> conflict-free
// broadcast reads: 16 distinct addresses in 16 distinct banks, 2 lanes each).
// ---------------------------------------------------------------------------
__global__ __launch_bounds__(256)
void pairwise_kernel(const float* __restrict__ Hq,
                     const float* __restrict__ Hk,
                     const float* __restrict__ sim,
                     const float* __restrict__ gw,
                     const float* __restrict__ consts,
                     const float* __restrict__ u,
                     float* __restrict__ out) {
    __shared__ float hqs[16 * 257];
    __shared__ float hks[16 * 257];
    __shared__ float gws[H_];
    const int tid = threadIdx.x;
    const int l0 = blockIdx.x * 16;
    const int s0 = blockIdx.y * 16;
    const int b  = blockIdx.z;
    const float* hq = Hq + ((size_t)b * L_ + l0) * H_;
    const float* hk = Hk + ((size_t)b * S_ + s0) * H_;
    for (int idx = tid; idx < 16 * H_; idx += 256) {
        int r = idx >> 8, c = idx & (H_ - 1);
        hqs[r * 257 + c] = hq[r * H_ + c];
        hks[r * 257 + c] = hk[r * H_ + c];
    }
    gws[tid] = gw[tid];
    __syncthreads();

    const int li = tid & 15, si = tid >> 4;
    const float* hqr = hqs + li * 257;
    const float* hkr = hks + si * 257;
    float s1 = 0.f, s2 = 0.f, s3 = 0.f;
#pragma unroll 8
    for (int h = 0; h < H_; ++h) {
        float x = hqr[h] + hkr[h];
        float g = 0.5f * x * (1.0f + erff(x * 0.7071067811865476f)); // exact GELU
        s1 += g;
        s2 = fmaf(g, g, s2);
        s3 = fmaf(g, gws[h], s3);
    }
    const float GWsum = consts[0];
    const float C1    = consts[1];
    const float mu   = s1 * (1.0f / H_);
    const float var  = fmaf(-mu, mu, s2 * (1.0f / H_));
    const float rstd = rsqrtf(var + 1e-5f);
    const float thr  = rstd * (s3 - mu * GWsum) + C1;

    const int l = l0 + li, s = s0 + si;
    const size_t pidx = ((size_t)b * L_ + l) * S_ + s;
    const float logits = (sim[pidx] - thr) * 0.1f; // /BASE_TEMP

    float u0 = u[pidx * 2 + 0], u1 = u[pidx * 2 + 1];
    u0 = fminf(fmaxf(u0, 1e-6f), 1.0f - 1e-6f);
    u1 = fminf(fmaxf(u1, 1e-6f), 1.0f - 1e-6f);
    const float g0 = -logf(-logf(u0));
    const float g1 = -logf(-logf(u1));
    // y = hard + soft - stop_grad(soft)  ==  hard numerically; argmax tie -> 0
    out[pidx] = ((logits + g1) > (-logits + g0)) ? 1.0f : 0.0f;
}

// ---------------------------------------------------------------------------
extern "C" void kernel_launch(void* const* d_in, const int* in_sizes, int n_in,
                              void* d_out, int out_size, void* d_ws, size_t ws_size,
                              hipStream_t stream) {
    const float* q     = (const float*)d_in[0];
    const float* k     = (const float*)d_in[1];
    const float* qg    = (const float*)d_in[2];
    const float* kg    = (const float*)d_in[3];
    const float* W1    = (const float*)d_in[4];
    const float* b1    = (const float*)d_in[5];
    const float* gamma = (const float*)d_in[6];
    const float* beta  = (const float*)d_in[7];
    const float* W2    = (const float*)d_in[8];
    const float* b2    = (const float*)d_in[9];
    const float* gu    = (const float*)d_in[10];
    float* out = (float*)d_out;

    float* ws     = (float*)d_ws;
    float* Hq     = ws;                       // 768*256
    float* Hk     = Hq + B_ * L_ * H_;        // 768*256
    float* simb   = Hk + B_ * S_ * H_;        // 2*384*384
    float* Gq     = simb + (size_t)B_ * L_ * S_;
    float* Gk     = Gq + B_ * H_;
    float* gw     = Gk + B_ * H_;
    float* consts = gw + H_;

    proj_globals_kernel<<<dim3(4), dim3(256), 0, stream>>>(qg, kg, W1, Gq, Gk);
    prep_kernel<<<dim3(1), dim3(256), 0, stream>>>(gamma, beta, W2, b2, gw, consts);

    // Hq = q@Wq + Gq + b1      (WMMA fp32, one wave per 16x16 tile)
    gemm_proj_kernel<<<dim3((B_ * L_) / 16, H_ / 16), dim3(32), 0, stream>>>(
        q, W1, Gq, b1, 1.0f, Hq);
    // Hk = k@Wk + Gk
    gemm_proj_kernel<<<dim3((B_ * S_) / 16, H_ / 16), dim3(32), 0, stream>>>(
        k, W1 + D_ * H_, Gk, b1, 0.0f, Hk);
    // sim = q@k^T / sqrt(D)
    gemm_sim_kernel<<<dim3(L_ / 16, S_ / 16, B_), dim3(32), 0, stream>>>(q, k, simb);

    // fused GELU + LayerNorm-threshold + Gumbel straight-through
    pairwise_kernel<<<dim3(L_ / 16, S_ / 16, B_), dim3(256), 0, stream>>>(
        Hq, Hk, simb, gw, consts, gu, out);
}